// VanillaQuantumLayer_90280212562548
// MI455X (gfx1250) — compile-verified
//
#include <hip/hip_runtime.h>

// ============================================================================
// VanillaQuantumLayer (4 qubits) — analytically reduced.
//
//   a_w   = tanh(x_w) * pi/2
//   z_w   = cos(theta_w)*cos(a_w) - cos(phi_w)*sin(theta_w)*sin(a_w)
//   out   = { z0, z0*z1, z0*z1*z2, z0*z1*z2*z3 }        (per row)
//
// (RY+Rot leave a product state; the CNOT chain is a basis permutation that
//  XOR-accumulates bits, so <Z_w> is the prefix product of per-qubit z_w.
//  omega drops out of |amp|^2.)
//
// Memory-bound streaming kernel: 16B NT load + 16B NT store per row.
// HBM floor @ 23.3 TB/s for 33.5 MB total traffic: ~1.4 us.
// ============================================================================

typedef float v4f __attribute__((ext_vector_type(4)));

#define HALF_PI 1.57079632679489661923f

__device__ __forceinline__ float fast_tanhf(float x) {
#if __has_builtin(__builtin_amdgcn_tanhf)
    // gfx1250 hardware transcendental v_tanh_f32
    return __builtin_amdgcn_tanhf(x);
#else
    // tanh(x) = 1 - 2/(exp(2x)+1); v_exp_f32 + v_rcp_f32.
    // Saturates correctly at +-1 for large |x| (exp -> inf/0).
    float e = __expf(2.0f * x);
    return 1.0f - 2.0f * __builtin_amdgcn_rcpf(e + 1.0f);
#endif
}

__device__ __forceinline__ float z_of(float x, float A, float B) {
    float a = fast_tanhf(x) * HALF_PI;          // a in (-pi/2, pi/2)
    return A * __cosf(a) - B * __sinf(a);       // v_cos_f32 / v_sin_f32
}

// One tiny block: fold params[4][3] into 8 uniform constants in workspace.
//   ab[0..3] = cos(theta_w)
//   ab[4..7] = cos(phi_w) * sin(theta_w)
// params ~ N(0,1): hardware v_sin/v_cos are exact enough; avoid the
// scalarized Payne-Hanek libm path (it was ~2000 SALU ops of dead weight).
__global__ void qc_setup(const float* __restrict__ params,
                         float* __restrict__ ab) {
    if (threadIdx.x == 0) {
#pragma unroll
        for (int w = 0; w < 4; ++w) {
            float phi = params[3 * w + 0];
            float th  = params[3 * w + 1];
            ab[w]     = __cosf(th);
            ab[4 + w] = __cosf(phi) * __sinf(th);
        }
    }
}

template <bool FROM_WS>
__global__ __launch_bounds__(256) void qc_main(const float* __restrict__ xin,
                                               const float* __restrict__ params,
                                               const float* __restrict__ ab,
                                               float* __restrict__ outp,
                                               int nrows) {
    float A[4], B[4];
    if constexpr (FROM_WS) {
        // Uniform address -> scalar loads
#pragma unroll
        for (int w = 0; w < 4; ++w) {
            A[w] = ab[w];
            B[w] = ab[4 + w];
        }
    } else {
        // Fallback if workspace is unusable: compute uniforms per-thread.
#pragma unroll
        for (int w = 0; w < 4; ++w) {
            float phi = params[3 * w + 0];
            float th  = params[3 * w + 1];
            A[w] = __cosf(th);
            B[w] = __cosf(phi) * __sinf(th);
        }
    }

    const v4f* __restrict__ x4 = (const v4f*)xin;
    v4f* __restrict__ o4       = (v4f*)outp;

    int i      = blockIdx.x * blockDim.x + threadIdx.x;
    int stride = gridDim.x * blockDim.x;

    for (; i < nrows; i += stride) {
        v4f xv = __builtin_nontemporal_load(x4 + i);   // global_load_b128 (NT)

        float z0 = z_of(xv.x, A[0], B[0]);
        float z1 = z_of(xv.y, A[1], B[1]);
        float z2 = z_of(xv.z, A[2], B[2]);
        float z3 = z_of(xv.w, A[3], B[3]);

        v4f o;
        o.x = z0;
        o.y = z0 * z1;
        o.z = o.y * z2;
        o.w = o.z * z3;

        __builtin_nontemporal_store(o, o4 + i);        // global_store_b128 (NT)
    }
}

extern "C" void kernel_launch(void* const* d_in, const int* in_sizes, int n_in,
                              void* d_out, int out_size, void* d_ws,
                              size_t ws_size, hipStream_t stream) {
    const float* x      = (const float*)d_in[0];   // [B, 4] float32
    const float* params = (const float*)d_in[1];   // [4, 3] float32
    float* out          = (float*)d_out;           // [B, 4] float32

    int nrows = in_sizes[0] / 4;                   // B = 1,048,576

    const int TPB  = 256;                          // 8 wave32 per workgroup
    const int ROWS = 4;                            // rows per thread (grid-stride)
    int blocks = (nrows + TPB * ROWS - 1) / (TPB * ROWS);
    if (blocks < 1) blocks = 1;

    if (ws_size >= 8 * sizeof(float)) {
        float* ab = (float*)d_ws;
        qc_setup<<<1, 32, 0, stream>>>(params, ab);
        qc_main<true><<<blocks, TPB, 0, stream>>>(x, params, ab, out, nrows);
    } else {
        qc_main<false><<<blocks, TPB, 0, stream>>>(x, params, nullptr, out, nrows);
    }
}